// EdgeDecoder_29171417874640
// MI455X (gfx1250) — compile-verified
//
#include <hip/hip_runtime.h>
#include <hip/hip_bf16.h>
#include <math.h>

typedef float v2f __attribute__((ext_vector_type(2)));
typedef float v8f __attribute__((ext_vector_type(8)));

#define D1 128
#define D2 64
#define WAVES_PER_BLOCK 8
#define BLOCK (WAVES_PER_BLOCK * 32)
#define ROW_STRIDE 132                 // floats per edge row in LDS (128 + 4 pad -> conflict-free A reads)
#define TILE_FLOATS (16 * ROW_STRIDE)  // 2112 floats per wave tile
#define W1_PAIRS (64 * 64)             // 4096 float2 (K-pair interleaved W1) = 32KB
#define LDS_FLOATS (2 * W1_PAIRS + WAVES_PER_BLOCK * TILE_FLOATS) // 8192 + 16896 = 25088 floats

__global__ void __launch_bounds__(BLOCK)
edge_decoder_kernel(const float* __restrict__ hf,   // [N_NODES, 128]
                    const int*   __restrict__ edge, // [2, E] int32
                    const float* __restrict__ W1,   // [128, 64]
                    const float* __restrict__ b1,   // [64]
                    const float* __restrict__ W2,   // [64, 1]
                    const float* __restrict__ b2,   // [1]
                    float* __restrict__ out,        // [E]
                    int E)
{
    extern __shared__ float smem[];
    v2f* W1p = (v2f*)smem;  // W1p[p*64 + c] = { W1[2p][c], W1[2p+1][c] }

    const int tid = threadIdx.x;

    // ---- Stage W1 into LDS, K-pair interleaved so a B fragment is one ds_load_b64 ----
    for (int idx = tid; idx < W1_PAIRS; idx += BLOCK) {
        const int p = idx >> 6;      // K-pair index (k = 2p, 2p+1)
        const int c = idx & 63;      // output column
        v2f v;
        v.x = W1[p * 2 * D2 + c];
        v.y = W1[p * 2 * D2 + D2 + c];
        W1p[idx] = v;
    }
    __syncthreads();

    const int l  = tid & 31;   // lane
    const int li = l & 15;     // lane within half
    const int hi = l >> 4;     // half select (0: lanes 0-15, 1: lanes 16-31)
    const int w  = tid >> 5;   // wave in block

    float* tileBuf = smem + 2 * W1_PAIRS + w * TILE_FLOATS;
    const v2f* Ab = (const v2f*)tileBuf;         // A-frag view (float2 units; row = 66 float2)
    const int aBase = li * (ROW_STRIDE / 2) + hi; // lanes 0-15: K=4t,4t+1 ; lanes 16-31: K=4t+2,4t+3

    const int* edge0 = edge;
    const int* edge1 = edge + E;

    // Per-lane loop invariants: bias / W2 entries for columns ncol = 16n + li
    float bias[4], w2v[4];
#pragma unroll
    for (int n = 0; n < 4; ++n) {
        bias[n] = b1[16 * n + li];
        w2v[n]  = W2[16 * n + li];
    }
    const float b2v = b2[0];

    const int ntiles = (E + 15) >> 4;
    const int nwaves = gridDim.x * WAVES_PER_BLOCK;

    for (int tile = blockIdx.x * WAVES_PER_BLOCK + w; tile < ntiles; tile += nwaves) {
        const int e0 = tile << 4;

        // ---- Gather both endpoints, elementwise multiply, stage [16 x 128] in LDS ----
#pragma unroll 4
        for (int e = 0; e < 16; ++e) {
            int ee = e0 + e; if (ee >= E) ee = E - 1;      // clamp for (unused here) tail tile
            const long long u = edge0[ee];
            const long long v = edge1[ee];
            const float4 fu = *(const float4*)(hf + u * D1 + 4 * l);
            const float4 fv = *(const float4*)(hf + v * D1 + 4 * l);
            float4 p;
            p.x = fu.x * fv.x; p.y = fu.y * fv.y;
            p.z = fu.z * fv.z; p.w = fu.w * fv.w;
            *(float4*)(tileBuf + e * ROW_STRIDE + 4 * l) = p;
        }
        // LDS ops from the same wave are issued/kept in order (ISA ch.11): A-frag
        // reads below see the stores above; compiler keeps order (may-alias LDS).

        // ---- [16x128] x [128x64] via 32 K-steps x 4 N-tiles of V_WMMA_F32_16X16X4_F32 ----
        v8f c0 = {bias[0],bias[0],bias[0],bias[0],bias[0],bias[0],bias[0],bias[0]};
        v8f c1 = {bias[1],bias[1],bias[1],bias[1],bias[1],bias[1],bias[1],bias[1]};
        v8f c2 = {bias[2],bias[2],bias[2],bias[2],bias[2],bias[2],bias[2],bias[2]};
        v8f c3 = {bias[3],bias[3],bias[3],bias[3],bias[3],bias[3],bias[3],bias[3]};

#pragma unroll
        for (int t = 0; t < 32; ++t) {
            const v2f a = Ab[aBase + 2 * t];
            const int bb = (2 * t + hi) * 64 + li;     // K-pair row (lo: 4t,4t+1 / hi: 4t+2,4t+3)
            c0 = __builtin_amdgcn_wmma_f32_16x16x4_f32(false, a, false, W1p[bb     ], (short)0, c0, false, false);
            c1 = __builtin_amdgcn_wmma_f32_16x16x4_f32(false, a, false, W1p[bb + 16], (short)0, c1, false, false);
            c2 = __builtin_amdgcn_wmma_f32_16x16x4_f32(false, a, false, W1p[bb + 32], (short)0, c2, false, false);
            c3 = __builtin_amdgcn_wmma_f32_16x16x4_f32(false, a, false, W1p[bb + 48], (short)0, c3, false, false);
        }

        // ---- ReLU + dot with W2 (column ncol = 16n + li), then reduce across the 16-lane half ----
        float s[8];
#pragma unroll
        for (int i = 0; i < 8; ++i) {
            s[i] = fmaxf(c0[i], 0.f) * w2v[0] + fmaxf(c1[i], 0.f) * w2v[1]
                 + fmaxf(c2[i], 0.f) * w2v[2] + fmaxf(c3[i], 0.f) * w2v[3];
        }
#pragma unroll
        for (int m = 1; m <= 8; m <<= 1) {
#pragma unroll
            for (int i = 0; i < 8; ++i) s[i] += __shfl_xor(s[i], m, 32);
        }

        // C/D layout: VGPR i -> M=i (lanes 0-15) / M=8+i (lanes 16-31). Lanes 0 & 16 store 8 edges each.
        if (li == 0) {
            const int base = e0 + hi * 8;
            float r[8];
#pragma unroll
            for (int i = 0; i < 8; ++i) r[i] = 1.f / (1.f + expf(-(s[i] + b2v)));
            if (base + 8 <= E) {
                float4 o0 = {r[0], r[1], r[2], r[3]};
                float4 o1 = {r[4], r[5], r[6], r[7]};
                *(float4*)(out + base)     = o0;
                *(float4*)(out + base + 4) = o1;
            } else {
#pragma unroll
                for (int i = 0; i < 8; ++i) if (base + i < E) out[base + i] = r[i];
            }
        }
    }
}

extern "C" void kernel_launch(void* const* d_in, const int* in_sizes, int n_in,
                              void* d_out, int out_size, void* d_ws, size_t ws_size,
                              hipStream_t stream) {
    (void)n_in; (void)out_size; (void)d_ws; (void)ws_size;
    const float* hf   = (const float*)d_in[0];
    const int*   edge = (const int*)d_in[1];
    const float* W1   = (const float*)d_in[2];
    const float* b1   = (const float*)d_in[3];
    const float* W2   = (const float*)d_in[4];
    const float* b2   = (const float*)d_in[5];
    float* out = (float*)d_out;

    const int E = in_sizes[1] / 2;
    const int ntiles = (E + 15) / 16;
    int blocks = (ntiles + WAVES_PER_BLOCK - 1) / WAVES_PER_BLOCK;
    if (blocks > 2500) blocks = 2500;   // grid-stride; amortizes the W1 LDS stage

    const size_t shmem = LDS_FLOATS * sizeof(float); // ~98 KB of the 320 KB WGP LDS
    hipLaunchKernelGGL(edge_decoder_kernel, dim3(blocks), dim3(BLOCK), shmem, stream,
                       hf, edge, W1, b1, W2, b2, out, E);
}